// OHEMBCEWithLogitsLoss_59957743452460
// MI455X (gfx1250) — compile-verified
//
#include <hip/hip_runtime.h>
#include <stdint.h>

// ---------------------------------------------------------------------------
// OHEM BCE-with-logits: mean of top-k losses, k = max(n/4, 100000).
// 2-level 16-bit radix select on raw float bits (loss >= 0 so the bit pattern
// is order-isomorphic). Key insight: elements in the same low-16 bin of the
// coarse bin B1 are bit-identical, so the tie-region sum comes straight from
// the histogram -- no third data pass.
// CDNA5 leverage: 320KB LDS/WGP -> full 65536-bin LDS histogram per WG,
// ds_add_u32/ds_add_u64 LDS atomics, b128 NT streaming loads, 192MB L2 to
// keep the 134MB loss cache resident, global_prefetch.
// ---------------------------------------------------------------------------

#define NBINS 65536
#define FIXSCALE 1073741824.0  // 2^30 fixed-point scale for u64 sums

typedef float v4f __attribute__((ext_vector_type(4)));

__device__ __forceinline__ float bce_loss(float x, float t) {
  // max(x,0) - x*t + log1p(exp(-|x|))  (numerically stable BCE-with-logits)
  return fmaxf(x, 0.0f) - x * t + log1pf(expf(-fabsf(x)));
}

// ---------------------------------------------------------------------------
// Zero histogram / control region (ws is poisoned 0xAA by the harness).
// ---------------------------------------------------------------------------
__global__ void zero_ws_kernel(uint32_t* __restrict__ p, int nwords) {
  int i = blockIdx.x * blockDim.x + threadIdx.x;
  if (i < nwords) p[i] = 0u;
}

// ---------------------------------------------------------------------------
// Pass A: compute loss, LDS histogram of top 16 bits of loss bit pattern.
// Inputs are read exactly once across the whole pipeline (cache mode), so
// load them non-temporally to avoid evicting the loss cache from L2.
// ---------------------------------------------------------------------------
__global__ __launch_bounds__(1024) void pass_hist_hi(
    const float* __restrict__ logits, const float* __restrict__ targets,
    float* __restrict__ lossCache, int useCache,
    uint32_t* __restrict__ hist, long n) {
  __shared__ uint32_t h[NBINS];  // 256 KB of the 320 KB/WGP LDS
  for (int i = threadIdx.x; i < NBINS; i += blockDim.x) h[i] = 0u;
  __syncthreads();

  long n4 = n >> 2;
  long stride = (long)gridDim.x * blockDim.x;
  long gid = (long)blockIdx.x * blockDim.x + threadIdx.x;

  const v4f* __restrict__ x4 = (const v4f*)logits;
  const v4f* __restrict__ t4 = (const v4f*)targets;
  v4f* __restrict__ c4 = (v4f*)lossCache;

  for (long i = gid; i < n4; i += stride) {
    __builtin_prefetch(&x4[i + stride], 0, 0);
    __builtin_prefetch(&t4[i + stride], 0, 0);
    v4f x = __builtin_nontemporal_load(&x4[i]);  // streamed, touched once
    v4f t = __builtin_nontemporal_load(&t4[i]);
    v4f l;
    l.x = bce_loss(x.x, t.x);
    l.y = bce_loss(x.y, t.y);
    l.z = bce_loss(x.z, t.z);
    l.w = bce_loss(x.w, t.w);
    atomicAdd(&h[__float_as_uint(l.x) >> 16], 1u);
    atomicAdd(&h[__float_as_uint(l.y) >> 16], 1u);
    atomicAdd(&h[__float_as_uint(l.z) >> 16], 1u);
    atomicAdd(&h[__float_as_uint(l.w) >> 16], 1u);
    if (useCache) c4[i] = l;  // regular store: keep resident in 192MB L2
  }
  long tail = n4 << 2;  // scalar tail if n % 4 != 0
  if (tail + gid < n) {
    float l = bce_loss(logits[tail + gid], targets[tail + gid]);
    atomicAdd(&h[__float_as_uint(l) >> 16], 1u);
    if (useCache) lossCache[tail + gid] = l;
  }

  __syncthreads();
  for (int i = threadIdx.x; i < NBINS; i += blockDim.x) {
    uint32_t c = h[i];
    if (c) atomicAdd(&hist[i], c);
  }
}

// ---------------------------------------------------------------------------
// Coarse scan: find bin B1 where cumulative-from-top crosses k.
// ctl[0] = B1, ctl[1] = count strictly above bin B1.
// ---------------------------------------------------------------------------
__global__ __launch_bounds__(1024) void scan_coarse(
    const uint32_t* __restrict__ hist, uint32_t* __restrict__ ctl, long k) {
  __shared__ uint32_t s[1024];
  int t = threadIdx.x;
  uint32_t sum = 0;
  const uint32_t* hp = hist + t * 64;
#pragma unroll
  for (int j = 0; j < 64; ++j) sum += hp[j];
  s[t] = sum;
  __syncthreads();

  if (t == 0) {
    unsigned long long kk = (unsigned long long)k;
    unsigned long long run = 0ull;
    for (int c = 1023; c >= 0; --c) {
      if (run + s[c] >= kk) {
        for (int b = c * 64 + 63;; --b) {
          uint32_t hc = hist[b];
          if (run + hc >= kk) {
            ctl[0] = (uint32_t)b;    // coarse bin B1
            ctl[1] = (uint32_t)run;  // count strictly above B1
            return;
          }
          run += hc;
        }
      }
      run += s[c];
    }
  }
}

// ---------------------------------------------------------------------------
// Pass B (fused): LDS histogram of low 16 bits for elements in bin B1, AND
// u64 fixed-point sum of all elements strictly above bin B1 (they are all
// guaranteed members of the top-k). Reads the L2-resident loss cache.
// ---------------------------------------------------------------------------
__global__ __launch_bounds__(1024) void pass_hist_lo_sum(
    const float* __restrict__ logits, const float* __restrict__ targets,
    const float* __restrict__ lossCache, int useCache,
    const uint32_t* __restrict__ ctl, uint32_t* __restrict__ hist,
    unsigned long long* __restrict__ gsum, long n) {
  __shared__ uint32_t h[NBINS];
  __shared__ unsigned long long bsum;
  for (int i = threadIdx.x; i < NBINS; i += blockDim.x) h[i] = 0u;
  if (threadIdx.x == 0) bsum = 0ull;
  __syncthreads();

  uint32_t b1 = ctl[0];
  long n4 = n >> 2;
  long stride = (long)gridDim.x * blockDim.x;
  long gid = (long)blockIdx.x * blockDim.x + threadIdx.x;

  const v4f* __restrict__ x4 = (const v4f*)logits;
  const v4f* __restrict__ t4 = (const v4f*)targets;
  const v4f* __restrict__ c4 = (const v4f*)lossCache;

  unsigned long long local = 0ull;
  for (long i = gid; i < n4; i += stride) {
    v4f l;
    if (useCache) {
      l = c4[i];  // hits L2
    } else {
      v4f x = __builtin_nontemporal_load(&x4[i]);
      v4f t = __builtin_nontemporal_load(&t4[i]);
      l.x = bce_loss(x.x, t.x);
      l.y = bce_loss(x.y, t.y);
      l.z = bce_loss(x.z, t.z);
      l.w = bce_loss(x.w, t.w);
    }
    uint32_t b[4] = {__float_as_uint(l.x), __float_as_uint(l.y),
                     __float_as_uint(l.z), __float_as_uint(l.w)};
    float lv[4] = {l.x, l.y, l.z, l.w};
#pragma unroll
    for (int e = 0; e < 4; ++e) {
      uint32_t hi = b[e] >> 16;
      if (hi == b1)
        atomicAdd(&h[b[e] & 0xFFFFu], 1u);
      else if (hi > b1)
        local += (unsigned long long)((double)lv[e] * FIXSCALE);
    }
  }
  long tail = n4 << 2;
  if (tail + gid < n) {
    float l = useCache ? lossCache[tail + gid]
                       : bce_loss(logits[tail + gid], targets[tail + gid]);
    uint32_t bb = __float_as_uint(l);
    uint32_t hi = bb >> 16;
    if (hi == b1)
      atomicAdd(&h[bb & 0xFFFFu], 1u);
    else if (hi > b1)
      local += (unsigned long long)((double)l * FIXSCALE);
  }

  if (local) atomicAdd(&bsum, local);  // ds_add_u64
  __syncthreads();
  for (int i = threadIdx.x; i < NBINS; i += blockDim.x) {
    uint32_t c = h[i];
    if (c) atomicAdd(&hist[i], c);
  }
  if (threadIdx.x == 0 && bsum) atomicAdd(gsum, bsum);
}

// ---------------------------------------------------------------------------
// Fine scan + finalize (fused). Elements in low-bin j of coarse bin B1 all
// have the bit-identical value float((B1<<16)|j), so the tie-region sum is
// sum(hist2[j] * value(j)) for bins above the crossing -- no data pass.
// mean = (sumAbove + sum_binsAboveV cnt*val + (k - cntG) * val(V)) / k
// ---------------------------------------------------------------------------
__global__ __launch_bounds__(1024) void scan_finalize(
    const uint32_t* __restrict__ hist, const uint32_t* __restrict__ ctl,
    const unsigned long long* __restrict__ gsum, float* __restrict__ out,
    long k) {
  __shared__ uint32_t s[1024];
  __shared__ double w[1024];
  int t = threadIdx.x;
  uint32_t baseBits = ctl[0] << 16;

  uint32_t cnt = 0;
  double wsum = 0.0;
#pragma unroll
  for (int j = 0; j < 64; ++j) {
    int b = t * 64 + j;
    uint32_t hc = hist[b];
    cnt += hc;
    if (hc) wsum += (double)hc * (double)__uint_as_float(baseBits | (uint32_t)b);
  }
  s[t] = cnt;
  w[t] = wsum;
  __syncthreads();

  if (t == 0) {
    unsigned long long kk = (unsigned long long)k;
    unsigned long long run = (unsigned long long)ctl[1];  // cntAbove
    double acc = 0.0;  // sum of tie-region bins strictly above threshold
    for (int c = 1023; c >= 0; --c) {
      if (run + s[c] >= kk) {
        for (int b = c * 64 + 63;; --b) {
          uint32_t hc = hist[b];
          if (run + hc >= kk) {
            double vV = (double)__uint_as_float(baseBits | (uint32_t)b);
            double total = (double)(*gsum) * (1.0 / FIXSCALE) + acc +
                           (double)(kk - run) * vV;
            out[0] = (float)(total / (double)k);
            return;
          }
          run += hc;
          acc += (double)hc * (double)__uint_as_float(baseBits | (uint32_t)b);
        }
      }
      run += s[c];
      acc += w[c];
    }
  }
}

// ---------------------------------------------------------------------------
extern "C" void kernel_launch(void* const* d_in, const int* in_sizes, int n_in,
                              void* d_out, int out_size, void* d_ws,
                              size_t ws_size, hipStream_t stream) {
  const float* logits = (const float*)d_in[0];
  const float* targets = (const float*)d_in[1];
  float* out = (float*)d_out;

  long n = (long)in_sizes[0];
  long k = (long)((double)n * 0.25);
  if (k < 100000) k = 100000;
  if (k > n) k = n;

  // Workspace layout:
  //   [0 .. 64K)   u32 hist1 (top-16-bit bins)
  //   [64K..128K)  u32 hist2 (low-16-bit bins, restricted to bin B1)
  //   [128K..+4)   u32 ctl[4]: B1, cntAbove, (spare), (spare)
  //   [+4 ..+6)    u64 gsum (8-byte aligned; fixed-point sum above bin B1)
  //   [1 MB .. )   optional f32 loss cache (n elements)
  uint32_t* ws32 = (uint32_t*)d_ws;
  uint32_t* hist1 = ws32;
  uint32_t* hist2 = ws32 + NBINS;
  uint32_t* ctl = ws32 + 2 * NBINS;
  unsigned long long* gsum = (unsigned long long*)(ws32 + 2 * NBINS + 4);
  float* lossCache = (float*)((char*)d_ws + (1 << 20));
  int useCache =
      (ws_size >= (size_t)(1 << 20) + (size_t)n * sizeof(float)) ? 1 : 0;

  int zeroWords = 2 * NBINS + 8;
  zero_ws_kernel<<<(zeroWords + 255) / 256, 256, 0, stream>>>(ws32, zeroWords);

  const int bigBlocks = 512;  // 1024 thr + 256KB LDS -> 32 waves on one WGP
  pass_hist_hi<<<bigBlocks, 1024, 0, stream>>>(logits, targets, lossCache,
                                               useCache, hist1, n);
  scan_coarse<<<1, 1024, 0, stream>>>(hist1, ctl, k);
  pass_hist_lo_sum<<<bigBlocks, 1024, 0, stream>>>(
      logits, targets, lossCache, useCache, ctl, hist2, gsum, n);
  scan_finalize<<<1, 1024, 0, stream>>>(hist2, ctl, gsum, out, k);
}